// KNRM_25847113187360
// MI455X (gfx1250) — compile-verified
//
#include <hip/hip_runtime.h>
#include <hip/hip_bf16.h>
#include <math.h>

// ---- KNRM: 21 kernels, mu = {1.0, 0.95, 0.85, ..., -0.95},
//      sigma = {0.001, 0.1 x20}.  Soft kernels are on an evenly spaced mu
//      grid (step 0.1), so consecutive Gaussians differ by a ratio
//      exp(-10m) * e^{9-j}: we evaluate the whole row with 5 exps instead
//      of 21 (transcendental pipe is the bottleneck at ~1/4 VALU rate).

typedef __attribute__((ext_vector_type(2))) float v2f;
typedef __attribute__((ext_vector_type(8))) float v8f;

#define NK 21
#define BQ 32      // query length
#define BD 1024    // doc length
#define EMB 128    // embedding dim
#define NCHUNK 8   // doc chunks per batch
#define CHUNK 128  // docs per chunk (2 passes of 64)
#define QSTRIDE 132 // LDS row stride (floats), 16B aligned, bank-conflict pad
#define MSTRIDE 17  // match LDS row stride

// Gaussian row for one match value m, accumulated into acc[21] with doc mask.
__device__ __forceinline__ void pool21(float m, float wm, float* acc) {
    // chain ratio constants: CL[j] = e^{9-j}, CR[j] = e^{j-9}
    constexpr float CL[19] = {
        8103.0839f,   2980.958f,    1096.6332f,   403.42879f,  148.41316f,
        54.59815f,    20.085537f,   7.3890562f,   2.7182818f,  1.0f,
        0.36787944f,  0.13533528f,  0.049787067f, 0.018315639f,
        0.0067379470f, 0.0024787523f, 0.00091188197f, 0.00033546262f,
        0.00012340980f};
    constexpr float CR[19] = {
        0.00012340980f, 0.00033546262f, 0.00091188197f, 0.0024787523f,
        0.0067379470f, 0.018315639f, 0.049787067f, 0.13533528f,
        0.36787944f,  1.0f,         2.7182818f,   7.3890562f,  20.085537f,
        54.59815f,    148.41316f,   403.42879f,   1096.6332f,  2980.958f,
        8103.0839f};

    // exact-match kernel (mu=1, sigma=1e-3): must stay a direct f32 exp
    const float d0 = m - 1.0f;
    acc[0] = fmaf(wm, __expf(-500000.0f * d0 * d0), acc[0]);

    const float t  = __expf(-10.0f * m);   // down-chain ratio factor
    const float ti = __expf( 10.0f * m);   // up-chain ratio factor
    const float dL = m - 0.95f;
    const float dR = m + 0.95f;

    float A[20];
    A[0] = __expf(-50.0f * dL * dL);       // seed at mu = +0.95 (k=1)
    #pragma unroll
    for (int j = 1; j < 20; ++j) A[j] = A[j - 1] * t * CL[j - 1];

    float bb = __expf(-50.0f * dR * dR);   // seed at mu = -0.95 (k=20)
    A[19] = fmaxf(A[19], bb);
    #pragma unroll
    for (int j = 18; j >= 0; --j) {
        bb = bb * ti * CR[j];
        A[j] = fmaxf(A[j], bb);            // each chain exact up to its peak
    }

    #pragma unroll
    for (int j = 0; j < 20; ++j) acc[j + 1] = fmaf(wm, A[j], acc[j + 1]);
}

// Fused: gather -> L2-normalize -> f32 WMMA GEMM -> gaussian kernel pooling.
// grid = (NCHUNK, B), block = 64 threads (2 waves of 32).
__global__ __launch_bounds__(64)
void knrm_main_kernel(const int* __restrict__ query,
                      const int* __restrict__ doc,
                      const float* __restrict__ d_mask,
                      const float* __restrict__ emb,
                      float* __restrict__ part) {
    const int b     = blockIdx.y;
    const int chunk = blockIdx.x;
    const int tid   = threadIdx.x;      // 0..63
    const int wave  = tid >> 5;         // 0..1  -> m-tile (query rows 16*wave..)
    const int lane  = tid & 31;

    __shared__ float qlds[BQ * QSTRIDE];      // normalized query rows
    __shared__ float dlds[64 * QSTRIDE];      // normalized doc rows (one pass)
    __shared__ float mlds[BQ * MSTRIDE];      // 32x16 match tile (both waves)
    __shared__ float dmask_s[64];
    __shared__ float accbuf[BQ * NK];

    // ---- Phase 0: gather + normalize the 32 query rows of this batch ----
    if (tid < BQ) {
        const int qid = query[b * BQ + tid];
        const float4* __restrict__ grow =
            (const float4*)(emb + (size_t)qid * EMB);
        float4* __restrict__ lrow = (float4*)(qlds + tid * QSTRIDE);
        float ss = 0.0f;
        #pragma unroll
        for (int i = 0; i < EMB / 4; ++i) {
            float4 v = grow[i];
            ss = fmaf(v.x, v.x, ss); ss = fmaf(v.y, v.y, ss);
            ss = fmaf(v.z, v.z, ss); ss = fmaf(v.w, v.w, ss);
            lrow[i] = v;
        }
        const float inv = 1.0f / fmaxf(sqrtf(ss), 1e-12f);
        #pragma unroll
        for (int i = 0; i < EMB / 4; ++i) {
            float4 v = lrow[i];
            v.x *= inv; v.y *= inv; v.z *= inv; v.w *= inv;
            lrow[i] = v;
        }
    }
    __syncthreads();

    float acc[NK];
    #pragma unroll
    for (int k = 0; k < NK; ++k) acc[k] = 0.0f;

    const int q  = tid & 31;   // query row this lane pools for
    const int dh = tid >> 5;   // doc half (0: cols 0..7, 1: cols 8..15)

    // A/B fragment addressing for V_WMMA_F32_16X16X4_F32:
    //   lanes 0-15: {K, K+1}, lanes 16-31: {K+2, K+3}; M(N) = lane & 15.
    const int ko = (lane >> 4) << 1;
    const float* __restrict__ arow = qlds + (wave * 16 + (lane & 15)) * QSTRIDE;

    for (int pass = 0; pass < 2; ++pass) {
        // ---- gather + normalize 64 doc rows into LDS (one row / thread) ----
        const int d   = chunk * CHUNK + pass * 64 + tid;
        const int did = doc[b * BD + d];
        {
            const float4* __restrict__ grow =
                (const float4*)(emb + (size_t)did * EMB);
            float4* __restrict__ lrow = (float4*)(dlds + tid * QSTRIDE);
            float ss = 0.0f;
            #pragma unroll
            for (int i = 0; i < EMB / 4; ++i) {
                float4 v = grow[i];
                ss = fmaf(v.x, v.x, ss); ss = fmaf(v.y, v.y, ss);
                ss = fmaf(v.z, v.z, ss); ss = fmaf(v.w, v.w, ss);
                lrow[i] = v;
            }
            const float inv = 1.0f / fmaxf(sqrtf(ss), 1e-12f);
            #pragma unroll
            for (int i = 0; i < EMB / 4; ++i) {
                float4 v = lrow[i];
                v.x *= inv; v.y *= inv; v.z *= inv; v.w *= inv;
                lrow[i] = v;
            }
            dmask_s[tid] = d_mask[b * BD + d];
        }
        __syncthreads();

        for (int dt = 0; dt < 4; ++dt) {
            // ---- 16x16 match tile via f32 WMMA, K = 128 in 32 steps ----
            const float* __restrict__ brow =
                dlds + (dt * 16 + (lane & 15)) * QSTRIDE;
            v8f c = {};
            #pragma unroll 8
            for (int ks = 0; ks < 32; ++ks) {
                const int kb = ks * 4 + ko;
                v2f a  = *(const v2f*)(arow + kb);
                v2f bb = *(const v2f*)(brow + kb);
                c = __builtin_amdgcn_wmma_f32_16x16x4_f32(
                        false, a, false, bb, (short)0, c, false, false);
            }
            // stash tile in LDS: lane holds col (lane&15), rows mrow0..mrow0+7
            const int mrow0 = wave * 16 + (lane >> 4) * 8;
            const int col   = lane & 15;
            #pragma unroll
            for (int r = 0; r < 8; ++r)
                mlds[(mrow0 + r) * MSTRIDE + col] = c[r];
            __syncthreads();

            // ---- gaussian kernel pooling (ratio-chain, 5 exps per match) ----
            #pragma unroll
            for (int j = 0; j < 8; ++j) {
                const int n  = dh * 8 + j;
                const float m  = mlds[q * MSTRIDE + n];
                const float wm = dmask_s[dt * 16 + n];
                pool21(m, wm, acc);
            }
            __syncthreads();
        }
    }

    // ---- combine the two doc-half accumulators, write deterministic partial
    if (dh == 1) {
        #pragma unroll
        for (int k = 0; k < NK; ++k) accbuf[q * NK + k] = acc[k];
    }
    __syncthreads();
    if (dh == 0) {
        float* __restrict__ o =
            part + (((size_t)b * NCHUNK + chunk) * BQ + q) * NK;
        #pragma unroll
        for (int k = 0; k < NK; ++k) o[k] = acc[k] + accbuf[q * NK + k];
    }
}

// log-pooling over q, dense layer, sigmoid. grid = B, block = 32.
__global__ __launch_bounds__(32)
void knrm_finalize_kernel(const float* __restrict__ part,
                          const float* __restrict__ q_mask,
                          const float* __restrict__ w_dense,
                          const float* __restrict__ b_dense,
                          float* __restrict__ out) {
    const int b = blockIdx.x;
    const int t = threadIdx.x;
    __shared__ float red[32];
    float s = 0.0f;
    if (t < NK) {
        for (int qq = 0; qq < BQ; ++qq) {
            float rs = 0.0f;
            #pragma unroll
            for (int c = 0; c < NCHUNK; ++c)
                rs += part[(((size_t)b * NCHUNK + c) * BQ + qq) * NK + t];
            s += __logf(fmaxf(rs, 1e-10f)) * q_mask[b * BQ + qq];
        }
        s *= 0.01f * w_dense[t];
    }
    red[t] = s;
    __syncthreads();
    if (t == 0) {
        float tot = b_dense[0];
        #pragma unroll
        for (int k = 0; k < NK; ++k) tot += red[k];
        out[b] = 1.0f / (1.0f + __expf(-tot));
    }
}

extern "C" void kernel_launch(void* const* d_in, const int* in_sizes, int n_in,
                              void* d_out, int out_size, void* d_ws, size_t ws_size,
                              hipStream_t stream) {
    const int*   query   = (const int*)d_in[0];   // (64,32)
    const int*   doc     = (const int*)d_in[1];   // (64,1024)
    const float* q_mask  = (const float*)d_in[2]; // (64,32)
    const float* d_mask  = (const float*)d_in[3]; // (64,1024)
    const float* emb     = (const float*)d_in[4]; // (100000,128)
    const float* w_dense = (const float*)d_in[5]; // (1,21)
    const float* b_dense = (const float*)d_in[6]; // (1,)
    float* out  = (float*)d_out;                  // (64,)
    float* part = (float*)d_ws;                   // (64,8,32,21) f32 partials

    dim3 grid(NCHUNK, 64);
    knrm_main_kernel<<<grid, 64, 0, stream>>>(query, doc, d_mask, emb, part);
    knrm_finalize_kernel<<<64, 32, 0, stream>>>(part, q_mask, w_dense, b_dense, out);
}